// MyAPPNP_28991029248859
// MI455X (gfx1250) — compile-verified
//
#include <hip/hip_runtime.h>

typedef __attribute__((ext_vector_type(2))) float v2f;
typedef __attribute__((ext_vector_type(8))) float v8f;

#define N_NODES 100000
#define F_IN    512
#define F_HID   256
#define F_OUT   64
#define N_EDGES 3200000
#define K_ITERS 10
#define ALPHA   0.1f

#define M_TILE  32               // rows per MLP block (8 waves)
#define XS (F_IN + 4)            // padded LDS stride for x tile
#define HS (F_HID + 4)           // padded LDS stride for h tile

// ---------------------------------------------------------------------------
// Fused MLP: x[32x512] -> h=relu(x@W1+b1)[32x256] -> x0=h@W2+b2[32x64]
// One block = 32 rows, 8 waves. Waves 0-3 handle rows 0-15, waves 4-7 rows
// 16-31; within a row-group each wave owns 64 h-columns (4 WMMA tiles).
// ---------------------------------------------------------------------------
__launch_bounds__(256)
__global__ void mlp_kernel(const float* __restrict__ x,
                           const float* __restrict__ W1,
                           const float* __restrict__ b1,
                           const float* __restrict__ W2,
                           const float* __restrict__ b2,
                           float* __restrict__ x0) {
  __shared__ float xs[M_TILE * XS];   // 66 KB
  __shared__ float hs[M_TILE * HS];   // 33.3 KB

  const int row0 = blockIdx.x * M_TILE;
  const int tid  = threadIdx.x;
  const int lane = tid & 31;
  const int wave = tid >> 5;
  const int mg   = wave >> 2;          // row group 0/1 -> rows 16*mg..16*mg+15

  // Cooperative load of the 32x512 x tile (float4, coalesced).
  for (int i = tid; i < M_TILE * (F_IN / 4); i += 256) {
    const int r = i / (F_IN / 4);
    const int c = (i % (F_IN / 4)) * 4;
    const float4 v = *(const float4*)(x + (size_t)(row0 + r) * F_IN + c);
    xs[r * XS + c + 0] = v.x;
    xs[r * XS + c + 1] = v.y;
    xs[r * XS + c + 2] = v.z;
    xs[r * XS + c + 3] = v.w;
  }
  __syncthreads();

  // A-operand lane mapping for V_WMMA_F32_16X16X4_F32:
  // lanes 0-15: M=lane, K=kb+{0,1}; lanes 16-31: M=lane-16, K=kb+{2,3}.
  const int m     = (lane & 15) + 16 * mg;   // LDS row for A reads
  const int koff  = (lane >> 4) * 2;         // 0 or 2
  const int nlane = lane & 15;
  const int mbase = (lane >> 4) * 8 + 16 * mg;  // C/D rows r / r+8 (+rowgroup)

  // ---- GEMM1: wave owns h columns [64*(wave&3), +64) as 4 tiles, K=512 ----
  v8f acc[4] = {v8f{}, v8f{}, v8f{}, v8f{}};
  const int nbase = (wave & 3) * 64;
  for (int kb = 0; kb < F_IN; kb += 4) {
    const v2f a = *(const v2f*)(&xs[m * XS + kb + koff]);
#pragma unroll
    for (int t = 0; t < 4; ++t) {
      const int n0 = nbase + t * 16;
      // B 4x16 tile: VGPR0 = rows kb+{0,2}, VGPR1 = rows kb+{1,3}
      const float* wp = W1 + (size_t)(kb + koff) * F_HID + n0 + nlane;
      v2f b;
      b.x = wp[0];
      b.y = wp[F_HID];
      acc[t] = __builtin_amdgcn_wmma_f32_16x16x4_f32(
          false, a, false, b, (short)0, acc[t], false, false);
    }
  }

  // Bias + ReLU epilogue -> LDS h tile.
#pragma unroll
  for (int t = 0; t < 4; ++t) {
    const int n0 = nbase + t * 16;
    const float bias = b1[n0 + nlane];
#pragma unroll
    for (int r = 0; r < 8; ++r) {
      float h = acc[t][r] + bias;
      h = h > 0.0f ? h : 0.0f;
      hs[(mbase + r) * HS + n0 + nlane] = h;
    }
  }
  __syncthreads();

  // ---- GEMM2: wave owns one 16-col tile of its row group, K=256 ----
  const int n0 = (wave & 3) * 16;
  v8f acc2 = v8f{};
  for (int kb = 0; kb < F_HID; kb += 4) {
    const v2f a = *(const v2f*)(&hs[m * HS + kb + koff]);
    const float* wp = W2 + (size_t)(kb + koff) * F_OUT + n0 + nlane;
    v2f b;
    b.x = wp[0];
    b.y = wp[F_OUT];
    acc2 = __builtin_amdgcn_wmma_f32_16x16x4_f32(
        false, a, false, b, (short)0, acc2, false, false);
  }

  const float bias2 = b2[n0 + nlane];
#pragma unroll
  for (int r = 0; r < 8; ++r) {
    const float v = acc2[r] + bias2;
    x0[(size_t)(row0 + mbase + r) * F_OUT + n0 + nlane] = v;
  }
}

// ---------------------------------------------------------------------------
// CSR build: zero, degree count, 3-phase exclusive scan, scatter
// ---------------------------------------------------------------------------
__launch_bounds__(256)
__global__ void zero_int_kernel(int* __restrict__ p, int n) {
  const int i = blockIdx.x * 256 + threadIdx.x;
  if (i < n) p[i] = 0;
}

__launch_bounds__(256)
__global__ void count_kernel(const int* __restrict__ ei, int* __restrict__ deg) {
  const int e = blockIdx.x * 256 + threadIdx.x;
  if (e < N_EDGES) atomicAdd(&deg[ei[e]], 1);
}

#define SCAN_BLOCK 256
__launch_bounds__(SCAN_BLOCK)
__global__ void scan_chunks_kernel(const int* __restrict__ deg,
                                   int* __restrict__ rptr,
                                   int* __restrict__ csums, int n) {
  __shared__ int sh[SCAN_BLOCK];
  const int i = blockIdx.x * SCAN_BLOCK + threadIdx.x;
  const int v = (i < n) ? deg[i] : 0;
  sh[threadIdx.x] = v;
  __syncthreads();
  for (int off = 1; off < SCAN_BLOCK; off <<= 1) {
    const int t = (threadIdx.x >= off) ? sh[threadIdx.x - off] : 0;
    __syncthreads();
    sh[threadIdx.x] += t;
    __syncthreads();
  }
  if (i < n) rptr[i] = sh[threadIdx.x] - v;          // exclusive
  if (threadIdx.x == SCAN_BLOCK - 1) csums[blockIdx.x] = sh[SCAN_BLOCK - 1];
}

__launch_bounds__(512)
__global__ void scan_sums_kernel(int* __restrict__ csums, int nc) {
  __shared__ int sh[512];
  const int v = (threadIdx.x < nc) ? csums[threadIdx.x] : 0;
  sh[threadIdx.x] = v;
  __syncthreads();
  for (int off = 1; off < 512; off <<= 1) {
    const int t = (threadIdx.x >= off) ? sh[threadIdx.x - off] : 0;
    __syncthreads();
    sh[threadIdx.x] += t;
    __syncthreads();
  }
  if (threadIdx.x < nc) csums[threadIdx.x] = sh[threadIdx.x] - v;  // exclusive
}

__launch_bounds__(SCAN_BLOCK)
__global__ void add_offsets_kernel(int* __restrict__ rptr,
                                   const int* __restrict__ csums,
                                   const int* __restrict__ deg, int n) {
  const int i = blockIdx.x * SCAN_BLOCK + threadIdx.x;
  if (i < n) {
    const int base = rptr[i] + csums[blockIdx.x];
    rptr[i] = base;
    if (i == n - 1) rptr[n] = base + deg[i];
  }
}

// CSR payload: one int2 {col, bitcast(w)} per edge -> single 8B load per edge
__launch_bounds__(256)
__global__ void scatter_kernel(const int* __restrict__ ei,
                               const float* __restrict__ ev,
                               const int* __restrict__ rptr,
                               int* __restrict__ fill,
                               int2* __restrict__ cpair) {
  const int e = blockIdx.x * 256 + threadIdx.x;
  if (e >= N_EDGES) return;
  const int row = ei[e];
  const int pos = rptr[row] + atomicAdd(&fill[row], 1);
  int2 p;
  p.x = ei[N_EDGES + e];
  p.y = __float_as_int(ev[e] * (1.0f - ALPHA));   // fold (1-alpha) into weight
  cpair[pos] = p;
}

// ---------------------------------------------------------------------------
// CSR SPMM: one wave32 per row. zdst[row] = ALPHA*x0[row] + sum_j w_j*z[col_j]
// row / rptr / (col,w) are wave-uniform: forced into SGPRs via readfirstlane
// so the per-edge loop is 1 uniform 8B load + saddr-form global_load_b64
// gather of z[col] (256 B contiguous, L2-resident) + 2 scalar-operand FMAs.
// ---------------------------------------------------------------------------
__launch_bounds__(256)
__global__ void spmm_csr_kernel(const int* __restrict__ rptr,
                                const int2* __restrict__ cpair,
                                const float* __restrict__ x0,
                                const float* __restrict__ zsrc,
                                float* __restrict__ zdst) {
  const int row =
      __builtin_amdgcn_readfirstlane((blockIdx.x * 256 + threadIdx.x) >> 5);
  const int lane = threadIdx.x & 31;
  if (row >= N_NODES) return;

  const int s = __builtin_amdgcn_readfirstlane(rptr[row]);
  const int e = __builtin_amdgcn_readfirstlane(rptr[row + 1]);

  __builtin_prefetch(&cpair[s], 0, 1);   // global_prefetch_b8 on edge stream

  const float2 xv = *(const float2*)(x0 + (size_t)row * F_OUT + lane * 2);
  float ax = xv.x * ALPHA;
  float ay = xv.y * ALPHA;

  for (int j = s; j < e; ++j) {
    const int2 p = cpair[j];                              // wave-uniform 8B
    const int   cb = __builtin_amdgcn_readfirstlane(p.x); // col  -> SGPR
    const float vb =
        __int_as_float(__builtin_amdgcn_readfirstlane(p.y));  // w -> SGPR
    const float2 zv = *(const float2*)(zsrc + (size_t)cb * F_OUT + lane * 2);
    ax = fmaf(vb, zv.x, ax);
    ay = fmaf(vb, zv.y, ay);
  }

  float2 o;
  o.x = ax;
  o.y = ay;
  *(float2*)(zdst + (size_t)row * F_OUT + lane * 2) = o;
}

// ---------------------------------------------------------------------------
extern "C" void kernel_launch(void* const* d_in, const int* in_sizes, int n_in,
                              void* d_out, int out_size, void* d_ws, size_t ws_size,
                              hipStream_t stream) {
  (void)in_sizes; (void)n_in; (void)out_size; (void)ws_size;

  const float* x  = (const float*)d_in[0];
  const float* W1 = (const float*)d_in[1];
  const float* b1 = (const float*)d_in[2];
  const float* W2 = (const float*)d_in[3];
  const float* b2 = (const float*)d_in[4];
  const float* ev = (const float*)d_in[5];
  const int*   ei = (const int*)d_in[6];
  float* out = (float*)d_out;

  // Workspace layout (16B-aligned slices)
  char* ws = (char*)d_ws;
  float* x0    = (float*)ws;                ws += (size_t)N_NODES * F_OUT * 4;
  float* zA    = (float*)ws;                ws += (size_t)N_NODES * F_OUT * 4;
  float* zB    = (float*)ws;                ws += (size_t)N_NODES * F_OUT * 4;
  int*   rptr  = (int*)ws;                  ws += ((size_t)N_NODES + 4) * 4;
  int*   deg   = (int*)ws;                  ws += (size_t)N_NODES * 4;
  int*   csum  = (int*)ws;                  ws += 512 * 4;
  int2*  cpair = (int2*)ws;

  const int eb  = (N_EDGES + 255) / 256;
  const int nb  = (N_NODES + 255) / 256;
  const int ncs = (N_NODES + SCAN_BLOCK - 1) / SCAN_BLOCK;   // 391 chunks

  // Fused MLP -> x0 (z_0 == x0, read-only thereafter)
  mlp_kernel<<<N_NODES / M_TILE, 256, 0, stream>>>(x, W1, b1, W2, b2, x0);

  // Build CSR (amortized over 10 propagation steps; rebuilt every call)
  zero_int_kernel<<<nb, 256, 0, stream>>>(deg, N_NODES);
  count_kernel<<<eb, 256, 0, stream>>>(ei, deg);
  scan_chunks_kernel<<<ncs, SCAN_BLOCK, 0, stream>>>(deg, rptr, csum, N_NODES);
  scan_sums_kernel<<<1, 512, 0, stream>>>(csum, ncs);
  add_offsets_kernel<<<ncs, SCAN_BLOCK, 0, stream>>>(rptr, csum, deg, N_NODES);
  zero_int_kernel<<<nb, 256, 0, stream>>>(deg, N_NODES);   // reuse as fill cursor
  scatter_kernel<<<eb, 256, 0, stream>>>(ei, ev, rptr, deg, cpair);

  // K propagation steps: z_{k+1} = ALPHA*x0 + sum_edges w * z_k[col]
  const int sb = ((size_t)N_NODES * 32 + 255) / 256;   // one wave per row
  for (int k = 0; k < K_ITERS; ++k) {
    const float* src = (k == 0) ? x0 : ((k & 1) ? zA : zB);
    float* dst = (k == K_ITERS - 1) ? out : ((k & 1) ? zB : zA);
    spmm_csr_kernel<<<sb, 256, 0, stream>>>(rptr, cpair, x0, src, dst);
  }
}